// CliffordInteractionExpert_89429809037614
// MI455X (gfx1250) — compile-verified
//
#include <hip/hip_runtime.h>

// CliffordInteractionExpert fused kernel for gfx1250 (MI455X).
//
// Math (derived from reference):
//   gate[b,t]  = sigmoid(x[b,t,:] . gw + gb)
//   S3[b,t,d]  = x[b,t-1,d] + x[b,t-2,d] + x[b,t-4,d]   (roll, wraps mod T)
//   sum_s scalar_s[b,t] = sum_d w16[d%16] * x_d * (3*x_d - S3_d)   (C0 is diagonal)
//   sum_s wedge_s at bivector k=i^j:  a_j*S3_i - a_i*S3_j          (wedge(a,a)=0)
//   out = x + gate * ( sb*wedgeSum  + [d==0] * ss*scalarSum )
//
// One wave = 16 consecutive t rows. lane -> (row = lane&15, col-half = lane>>4).
// Cross-lane per-row reductions done with V_WMMA_F32_16X16X4_F32:
//   A=(partial,0) per lane, B=ones -> D[i][j] = p_i + p_{i+16} (exact fp32).

typedef __attribute__((ext_vector_type(2))) float v2f;
typedef __attribute__((ext_vector_type(4))) float v4f;
typedef __attribute__((ext_vector_type(8))) float v8f;

#define T_DIM 4096
#define D_DIM 1024

__device__ __forceinline__ void load16(const float* __restrict__ p, float* a) {
#pragma unroll
  for (int q = 0; q < 4; ++q) {
    const v4f v = *(const v4f*)(p + 4 * q);
    a[4 * q + 0] = v[0]; a[4 * q + 1] = v[1];
    a[4 * q + 2] = v[2]; a[4 * q + 3] = v[3];
  }
}

__global__ __launch_bounds__(128) void clifford_fused(
    const float* __restrict__ x, const float* __restrict__ gw,
    const float* __restrict__ gb, const float* __restrict__ sw,
    const float* __restrict__ bw, float* __restrict__ out) {
  const int lane = threadIdx.x & 31;
  const int wave = threadIdx.x >> 5;
  const int tile = blockIdx.x * 4 + wave;   // 4 waves per block
  const int b  = tile >> 8;                 // T/16 = 256 tiles per batch
  const int tg = tile & 255;
  const int r  = lane & 15;                 // row within 16-row tile
  const int h  = lane >> 4;                 // column-half
  const int t  = tg * 16 + r;

  const size_t base = ((size_t)b * T_DIM + t) * D_DIM;
  const size_t bm1  = ((size_t)b * T_DIM + ((t - 1) & (T_DIM - 1))) * D_DIM;
  const size_t bm2  = ((size_t)b * T_DIM + ((t - 2) & (T_DIM - 1))) * D_DIM;
  const size_t bm4  = ((size_t)b * T_DIM + ((t - 4) & (T_DIM - 1))) * D_DIM;

  // diag(C0) for metric [+,+,+,-]:  C[a,a,0], a = 0..15
  const float W16[16] = {1.f, 1.f, 1.f, -1.f,  1.f, -1.f, -1.f, -1.f,
                         -1.f, 1.f, 1.f,  1.f,  1.f,  1.f,  1.f, -1.f};

  // ---------------- Phase 1: per-row reductions (gate logit, combined scalar)
  float gp = 0.f, scp = 0.f;
  for (int j = 0; j < 32; ++j) {
    const int cb = (2 * j + h) << 4;        // this lane's 16-wide chunk base
    float a[16], g[16], m1[16], m2[16], m4[16];
    load16(x + base + cb, a);
    load16(gw + cb, g);
    load16(x + bm1 + cb, m1);
    load16(x + bm2 + cb, m2);
    load16(x + bm4 + cb, m4);
    if (j < 30) __builtin_prefetch(x + base + cb + 128, 0, 1);  // global_prefetch
#pragma unroll
    for (int m = 0; m < 16; ++m) {
      gp = fmaf(a[m], g[m], gp);
      const float S  = m1[m] + m2[m] + m4[m];
      const float t3 = fmaf(3.f, a[m], -S);
      scp = fmaf(W16[m] * a[m], t3, scp);
    }
  }

  // WMMA cross-lane reduction: D[i][j] = p(lane i) + p(lane i+16), all j.
  // (B = all-ones is invariant to K-row striping; exact fp32.)
  const v2f Bones = {1.f, 1.f};
  const v8f Cz = {};
  v2f Aa = {gp, 0.f};
  v8f Dg = __builtin_amdgcn_wmma_f32_16x16x4_f32(
      false, Aa, false, Bones, (short)0, Cz, false, false);
  Aa[0] = scp;
  v8f Ds = __builtin_amdgcn_wmma_f32_16x16x4_f32(
      false, Aa, false, Bones, (short)0, Cz, false, false);

  // Broadcast row totals through LDS: C/D layout -> lane 0 holds rows 0-7
  // (one per VGPR), lane 16 holds rows 8-15.  Plain __shared__ + barrier so
  // this lowers to ds_store_b32/ds_load_b32 (not sys-scope flat ops).
  __shared__ float red[4][2][16];
  if (lane == 0) {
#pragma unroll
    for (int k = 0; k < 8; ++k) { red[wave][0][k] = Dg[k]; red[wave][1][k] = Ds[k]; }
  }
  if (lane == 16) {
#pragma unroll
    for (int k = 0; k < 8; ++k) { red[wave][0][8 + k] = Dg[k]; red[wave][1][8 + k] = Ds[k]; }
  }
  __syncthreads();
  const float logit = red[wave][0][r] + gb[0];
  const float scSum = red[wave][1][r];

  const float gate = 1.f / (1.f + expf(-logit));
  const float ssig = 1.f / (1.f + expf(-sw[0]));
  const float bsig = 1.f / (1.f + expf(-bw[0]));
  const float gss = gate * ssig;
  const float gsb = gate * bsig;

  // ---------------- Phase 2: wedge (6 bivectors/chunk) + scalar at d==0, store
  for (int j = 0; j < 32; ++j) {
    const int cb = (2 * j + h) << 4;
    float a[16];
    load16(x + base + cb, a);
    // Only vector components {1,2,4,8} of S3 are needed here.
    const float S1 = x[bm1 + cb + 1] + x[bm2 + cb + 1] + x[bm4 + cb + 1];
    const float S2 = x[bm1 + cb + 2] + x[bm2 + cb + 2] + x[bm4 + cb + 2];
    const float S4 = x[bm1 + cb + 4] + x[bm2 + cb + 4] + x[bm4 + cb + 4];
    const float S8 = x[bm1 + cb + 8] + x[bm2 + cb + 8] + x[bm4 + cb + 8];

    a[3]  = fmaf(gsb, fmaf(a[2], S1, -(a[1] * S2)), a[3]);   // e0^e1
    a[5]  = fmaf(gsb, fmaf(a[4], S1, -(a[1] * S4)), a[5]);   // e0^e2
    a[6]  = fmaf(gsb, fmaf(a[4], S2, -(a[2] * S4)), a[6]);   // e1^e2
    a[9]  = fmaf(gsb, fmaf(a[8], S1, -(a[1] * S8)), a[9]);   // e0^e3
    a[10] = fmaf(gsb, fmaf(a[8], S2, -(a[2] * S8)), a[10]);  // e1^e3
    a[12] = fmaf(gsb, fmaf(a[8], S4, -(a[4] * S8)), a[12]);  // e2^e3
    if (cb == 0) a[0] = fmaf(gss, scSum, a[0]);              // global d==0 only

#pragma unroll
    for (int q = 0; q < 4; ++q) {
      v4f v;
      v[0] = a[4 * q + 0]; v[1] = a[4 * q + 1];
      v[2] = a[4 * q + 2]; v[3] = a[4 * q + 3];
      __builtin_nontemporal_store(v, (v4f*)(out + base + cb + 4 * q));
    }
  }
}

extern "C" void kernel_launch(void* const* d_in, const int* in_sizes, int n_in,
                              void* d_out, int out_size, void* d_ws, size_t ws_size,
                              hipStream_t stream) {
  (void)n_in; (void)out_size; (void)d_ws; (void)ws_size;
  const float* x  = (const float*)d_in[0];
  const float* gw = (const float*)d_in[1];
  const float* gb = (const float*)d_in[2];
  const float* sw = (const float*)d_in[3];
  const float* bw = (const float*)d_in[4];
  float* out = (float*)d_out;

  const int B = in_sizes[0] / (T_DIM * D_DIM);      // 8
  const int tiles = B * (T_DIM / 16);               // waves (16 rows each)
  dim3 grid(tiles / 4);                             // 4 waves / block
  dim3 block(128);
  clifford_fused<<<grid, block, 0, stream>>>(x, gw, gb, sw, bw, out);
}